// PortfolioModel_50757923504734
// MI455X (gfx1250) — compile-verified
//
#include <hip/hip_runtime.h>

#ifndef PDIM
#define PDIM 5
#endif
#define UBV 0.3f
#define RFV 1.0001f
#define NIT 24

typedef float v2f __attribute__((ext_vector_type(2)));
typedef float v8f __attribute__((ext_vector_type(8)));

// ds_swizzle group-of-32 encoding: and=0x1f, or=0, xor=0x10 -> lane <-> lane^16
__device__ __forceinline__ float swapx16(float x) {
    int i = __builtin_amdgcn_ds_swizzle(__builtin_bit_cast(int, x), 0x401f);
    return __builtin_bit_cast(float, i);
}

__device__ __forceinline__ float fast_tanh(float x) {
#if __has_builtin(__builtin_amdgcn_tanhf)
    return __builtin_amdgcn_tanhf(x);
#else
    float r;
    // CDNA5 trans op; v_nop covers the 1-op trans hazard window
    asm("v_tanh_f32 %0, %1\n\tv_nop" : "=v"(r) : "v"(x));
    return r;
#endif
}

// A-operand registers for one (zero-padded 16x8) weight matrix, per lane.
// Lane m (0-15): k-tile0 holds W[m][0],W[m][1]; lane m+16: W[m][2],W[m][3].
// k-tile1 holds W[m][4] in lanes 0-15 only (K=5..7 are zero padding).
struct WOp { v2f k0; v2f k1; };

__device__ __forceinline__ WOp make_wop(const float* __restrict__ W, int m, bool lo) {
    WOp w;
    if (m < PDIM) {
        w.k0.x = W[m * PDIM + (lo ? 0 : 2)];
        w.k0.y = W[m * PDIM + (lo ? 1 : 3)];
        w.k1.x = lo ? W[m * PDIM + 4] : 0.0f;
    } else {
        w.k0.x = 0.0f; w.k0.y = 0.0f; w.k1.x = 0.0f;
    }
    w.k1.y = 0.0f;
    return w;
}

// h (lane = row, h[0..4] per lane, 32 rows per wave) -> h = h @ W^T via
// D[m][n] = sum_k W[m][k] * h_rown[k], two 16-row tiles x two K-tiles.
__device__ __forceinline__ void matmul5(float h[PDIM], const WOp& w, bool lo) {
    float s0 = swapx16(h[0]), s1 = swapx16(h[1]), s2 = swapx16(h[2]),
          s3 = swapx16(h[3]), s4 = swapx16(h[4]);
    v2f b1k0, b1k1, b2k0, b2k1;
    // row-tile 1 (rows 0-15): lane n -> K0,K1 of row n ; lane n+16 -> K2,K3 of row n
    b1k0.x = lo ? h[0] : s2;   b1k0.y = lo ? h[1] : s3;
    b1k1.x = lo ? h[4] : 0.0f; b1k1.y = 0.0f;
    // row-tile 2 (rows 16-31): lane n -> K0,K1 of row 16+n ; lane n+16 -> K2,K3
    b2k0.x = lo ? s0 : h[2];   b2k0.y = lo ? s1 : h[3];
    b2k1.x = lo ? s4 : 0.0f;   b2k1.y = 0.0f;

    v8f c = {0.0f, 0.0f, 0.0f, 0.0f, 0.0f, 0.0f, 0.0f, 0.0f};
    v8f d1 = __builtin_amdgcn_wmma_f32_16x16x4_f32(false, w.k0, false, b1k0, (short)0, c,  false, false);
    d1     = __builtin_amdgcn_wmma_f32_16x16x4_f32(false, w.k1, false, b1k1, (short)0, d1, false, false);
    v8f d2 = __builtin_amdgcn_wmma_f32_16x16x4_f32(false, w.k0, false, b2k0, (short)0, c,  false, false);
    d2     = __builtin_amdgcn_wmma_f32_16x16x4_f32(false, w.k1, false, b2k1, (short)0, d2, false, false);

    // Merge back to lane=row: lanes<16 take d1 (rows 0-15), lanes>=16 take
    // d2 swizzled across halves (row 16+n computed in lane n of d2).
#pragma unroll
    for (int j = 0; j < PDIM; ++j) {
        float hi = swapx16(d2[j]);
        h[j] = lo ? d1[j] : hi;
    }
}

// softmax + 24-iter rebalance + wealth + stores (shared by WMMA and tail paths)
__device__ __forceinline__ void epilogue(const float* __restrict__ x1,
                                         const float h[PDIM], const float R[PDIM],
                                         float* __restrict__ out, int nrows, int row) {
    float mx = h[0];
#pragma unroll
    for (int j = 1; j < PDIM; ++j) mx = fmaxf(mx, h[j]);
    float e[PDIM], s = 0.0f;
#pragma unroll
    for (int j = 0; j < PDIM; ++j) { e[j] = __expf(h[j] - mx); s += e[j]; }
    float inv = 1.0f / s;

    float ov[PDIM], nw[PDIM];
#pragma unroll
    for (int j = 0; j < PDIM; ++j) {
        float w = e[j] * inv;
        ov[j] = w;
        nw[j] = fminf(fmaxf(w, 0.0f), UBV);
    }
    for (int it = 0; it < NIT; ++it) {
        float lf = 0.0f;
#pragma unroll
        for (int j = 0; j < PDIM; ++j) lf += ov[j] - nw[j];
        float dn = 0.0f;
        bool mk[PDIM];
#pragma unroll
        for (int j = 0; j < PDIM; ++j) { mk[j] = (nw[j] != UBV); if (mk[j]) dn += nw[j]; }
        dn = (dn == 0.0f) ? 1.0f : dn;
        float r = lf / dn;
        float n2[PDIM];
        bool any = false;
#pragma unroll
        for (int j = 0; j < PDIM; ++j) {
            n2[j] = nw[j] + (mk[j] ? r * nw[j] : 0.0f);
            any = any || (n2[j] > UBV);
        }
#pragma unroll
        for (int j = 0; j < PDIM; ++j) {
            float cl = fminf(fmaxf(n2[j], 0.0f), UBV);
            nw[j] = any ? cl : n2[j];
            ov[j] = n2[j];
        }
    }
    float wl = 0.0f;
#pragma unroll
    for (int j = 0; j < PDIM; ++j) wl += x1[row * PDIM + j] * nw[j] * (1.0f + R[j]);
    wl *= RFV;
    out[row] = wl;
#pragma unroll
    for (int j = 0; j < PDIM; ++j) out[nrows + row * PDIM + j] = R[j];
}

__global__ __launch_bounds__(256) void portfolio_wmma_kernel(
    const float* __restrict__ x1, const float* __restrict__ x2,
    const float* __restrict__ eps, const float* __restrict__ alpha,
    const float* __restrict__ Amat, const float* __restrict__ W1,
    const float* __restrict__ W2, const float* __restrict__ W3,
    float* __restrict__ out, int nrows) {
    const int lane = threadIdx.x & 31;
    const bool lo = lane < 16;
    const int m = lane & 15;
    const int row = blockIdx.x * blockDim.x + threadIdx.x; // wave = 32 consecutive rows

    WOp wA = make_wop(Amat, m, lo);
    WOp w1 = make_wop(W1, m, lo);
    WOp w2 = make_wop(W2, m, lo);
    WOp w3 = make_wop(W3, m, lo);

    float h[PDIM], R[PDIM];
#pragma unroll
    for (int j = 0; j < PDIM; ++j) h[j] = x2[row * PDIM + j];

    matmul5(h, wA, lo);                     // x2 @ A^T
#pragma unroll
    for (int j = 0; j < PDIM; ++j) {
        R[j] = h[j] + alpha[j] + eps[row * PDIM + j];
        h[j] = R[j];
    }
    matmul5(h, w1, lo);
#pragma unroll
    for (int j = 0; j < PDIM; ++j) h[j] = fast_tanh(h[j]);
    matmul5(h, w2, lo);
#pragma unroll
    for (int j = 0; j < PDIM; ++j) h[j] = fast_tanh(h[j]);
    matmul5(h, w3, lo);
#pragma unroll
    for (int j = 0; j < PDIM; ++j) h[j] = fast_tanh(h[j]);

    epilogue(x1, h, R, out, nrows, row);
}

// Scalar-FMA tail for any rows beyond a multiple of 256 (WMMA needs EXEC==all-1s,
// so the tail avoids WMMA entirely).
__device__ __forceinline__ void matvec5(float h[PDIM], const float* __restrict__ W) {
    float o[PDIM];
#pragma unroll
    for (int mm = 0; mm < PDIM; ++mm) {
        float a = 0.0f;
#pragma unroll
        for (int k = 0; k < PDIM; ++k) a = fmaf(W[mm * PDIM + k], h[k], a);
        o[mm] = a;
    }
#pragma unroll
    for (int j = 0; j < PDIM; ++j) h[j] = o[j];
}

__global__ __launch_bounds__(256) void portfolio_tail_kernel(
    const float* __restrict__ x1, const float* __restrict__ x2,
    const float* __restrict__ eps, const float* __restrict__ alpha,
    const float* __restrict__ Amat, const float* __restrict__ W1,
    const float* __restrict__ W2, const float* __restrict__ W3,
    float* __restrict__ out, int nrows, int start) {
    int row = start + blockIdx.x * blockDim.x + threadIdx.x;
    if (row >= nrows) return;
    float h[PDIM], R[PDIM];
#pragma unroll
    for (int j = 0; j < PDIM; ++j) h[j] = x2[row * PDIM + j];
    matvec5(h, Amat);
#pragma unroll
    for (int j = 0; j < PDIM; ++j) { R[j] = h[j] + alpha[j] + eps[row * PDIM + j]; h[j] = R[j]; }
    matvec5(h, W1);
#pragma unroll
    for (int j = 0; j < PDIM; ++j) h[j] = fast_tanh(h[j]);
    matvec5(h, W2);
#pragma unroll
    for (int j = 0; j < PDIM; ++j) h[j] = fast_tanh(h[j]);
    matvec5(h, W3);
#pragma unroll
    for (int j = 0; j < PDIM; ++j) h[j] = fast_tanh(h[j]);
    epilogue(x1, h, R, out, nrows, row);
}

extern "C" void kernel_launch(void* const* d_in, const int* in_sizes, int n_in,
                              void* d_out, int out_size, void* d_ws, size_t ws_size,
                              hipStream_t stream) {
    const float* x1    = (const float*)d_in[0];
    const float* x2    = (const float*)d_in[1];
    const float* eps   = (const float*)d_in[2];
    const float* alpha = (const float*)d_in[3];
    const float* Amat  = (const float*)d_in[4];
    const float* W1    = (const float*)d_in[5];
    const float* W2    = (const float*)d_in[6];
    const float* W3    = (const float*)d_in[7];
    float* out = (float*)d_out;

    const int nrows = in_sizes[0] / PDIM;   // B = 2097152
    const int block = 256;
    const int grid  = nrows / block;        // exact for B = 2^21

    if (grid > 0) {
        hipLaunchKernelGGL(portfolio_wmma_kernel, dim3(grid), dim3(block), 0, stream,
                           x1, x2, eps, alpha, Amat, W1, W2, W3, out, nrows);
    }
    const int done = grid * block;
    const int tail = nrows - done;
    if (tail > 0) {
        hipLaunchKernelGGL(portfolio_tail_kernel, dim3((tail + block - 1) / block),
                           dim3(block), 0, stream,
                           x1, x2, eps, alpha, Amat, W1, W2, W3, out, nrows, done);
    }
}